// BipartiteRemap_77189152244014
// MI455X (gfx1250) — compile-verified
//
#include <hip/hip_runtime.h>

typedef __attribute__((ext_vector_type(2))) float v2f;
typedef __attribute__((ext_vector_type(8))) float v8f;

#define N_SRC   100000
#define N_OUT   100000
#define N_EDGES 1600000
#define C       128

// workspace layout in floats
#define WX_OFF  0
#define S_OFF   12800000
#define NUM_OFF 12900000
#define DEN_OFF 25700000

// ---------------------------------------------------------------------------
// Phase 1: Wx[n][o] = sum_c x[c][n] * W[o][c] + b[o]   via V_WMMA_F32_16X16X4_F32
// One wave computes a 16-node x 128-channel strip. A (x) fragments preloaded
// into registers (x read exactly once); W streamed from cache (64 KB, L0/L2
// resident). 8 output tiles x 32 k-steps = 256 WMMAs per wave.
// ---------------------------------------------------------------------------
__global__ __launch_bounds__(128) void gat_gemm_wx(const float* __restrict__ x,
                                                   const float* __restrict__ W,
                                                   const float* __restrict__ b,
                                                   float* __restrict__ Wx) {
    const int lane = threadIdx.x & 31;
    const int wave = threadIdx.x >> 5;
    const int n0   = blockIdx.x * 64 + wave * 16;   // 4 waves * 16 nodes per block
    const int m    = lane & 15;                     // M (or N) index within tile
    const int kb   = (lane >> 4) * 2;               // K sub-index from lane half
    const int half = lane >> 4;

    // clamp node index for loads (WMMA needs full EXEC; no divergent loads)
    int n = n0 + m;
    if (n > N_SRC - 1) n = N_SRC - 1;

    // Preload all A fragments: A[m][k] = x[k][n] for K = 0..127 in steps of 4.
    v2f a[32];
#pragma unroll
    for (int k4 = 0; k4 < 32; ++k4) {
        const size_t c = (size_t)(k4 * 4 + kb);
        a[k4].x = x[c * N_SRC + n];
        a[k4].y = x[(c + 1) * N_SRC + n];
    }

    // 8 output-channel tiles of 16
    for (int t = 0; t < 8; ++t) {
        const int o0 = t * 16;
        v8f acc = {};
#pragma unroll
        for (int k4 = 0; k4 < 32; ++k4) {
            const int c = k4 * 4 + kb;
            v2f bf;
            bf.x = W[(o0 + m) * C + c];       // B[k][o] = W[o][c]
            bf.y = W[(o0 + m) * C + c + 1];
            acc = __builtin_amdgcn_wmma_f32_16x16x4_f32(
                false, a[k4], false, bf, (short)0, acc, false, false);
        }
        const float bias = b[o0 + m];
        // D layout: VGPR v -> row M = v + 8*half, column N = m
#pragma unroll
        for (int v = 0; v < 8; ++v) {
            const int nn = n0 + v + half * 8;
            if (nn < N_SRC) Wx[(size_t)nn * C + o0 + m] = acc[v] + bias;
        }
    }
}

// ---------------------------------------------------------------------------
// Phase 2a: per-node scalar s[n] = exp(lrelu(Wx[n] . att)). One wave per node,
// each lane handles a float4 chunk (32 x 4 = 128 channels), wave32 reduction.
// ---------------------------------------------------------------------------
__global__ __launch_bounds__(256) void gat_node_scalar(const float* __restrict__ Wx,
                                                       const float* __restrict__ att,
                                                       const float* __restrict__ alpha_p,
                                                       float* __restrict__ s) {
    const int gw   = (int)((blockIdx.x * blockDim.x + threadIdx.x) >> 5);
    const int lane = threadIdx.x & 31;
    if (gw >= N_SRC) return;

    const float4 wv = ((const float4*)(Wx + (size_t)gw * C))[lane];
    const float4 av = ((const float4*)att)[lane];
    float r = wv.x * av.x + wv.y * av.y + wv.z * av.z + wv.w * av.w;
#pragma unroll
    for (int off = 16; off > 0; off >>= 1) r += __shfl_xor(r, off, 32);
    if (lane == 0) {
        const float alpha = alpha_p[0];
        const float aa = (r >= 0.f) ? r : alpha * r;
        s[gw] = expf(aa);
    }
}

// ---------------------------------------------------------------------------
// Phase 2b: edge scatter. One wave per edge: lane l gathers float4 of Wx[src],
// scales by s[src], atomically accumulates into num[tgt]; lane 0 adds den.
// Wx and num are L2-resident (51.2 MB each, 192 MB L2) -> L2-atomic bound.
// ---------------------------------------------------------------------------
__global__ __launch_bounds__(256) void gat_edge_scatter(const int* __restrict__ edges,
                                                        const float* __restrict__ Wx,
                                                        const float* __restrict__ s,
                                                        float* __restrict__ num,
                                                        float* __restrict__ den) {
    const int gw   = (int)((blockIdx.x * blockDim.x + threadIdx.x) >> 5);
    const int lane = threadIdx.x & 31;
    if (gw >= N_EDGES) return;

    const int tgt = edges[2 * gw + 0];
    const int src = edges[2 * gw + 1];
    const float sv = s[src];
    const float4 p = ((const float4*)(Wx + (size_t)src * C))[lane];
    float* dst = num + (size_t)tgt * C + lane * 4;
    atomicAdd(dst + 0, p.x * sv);
    atomicAdd(dst + 1, p.y * sv);
    atomicAdd(dst + 2, p.z * sv);
    atomicAdd(dst + 3, p.w * sv);
    if (lane == 0) atomicAdd(den + tgt, sv);
}

// ---------------------------------------------------------------------------
// Phase 3: y[o][i] = num[i][o] / (den[i]==0 ? 1 : den[i]); LDS-tiled transpose
// so both the num read and the y write are coalesced.
// ---------------------------------------------------------------------------
__global__ __launch_bounds__(256) void gat_divide_transpose(const float* __restrict__ num,
                                                            const float* __restrict__ den,
                                                            float* __restrict__ y) {
    __shared__ float tile[32][33];
    const int ocol = blockIdx.x * 32;   // channel tile (4 tiles of 32)
    const int nrow = blockIdx.y * 32;   // node tile
#pragma unroll
    for (int j = 0; j < 4; ++j) {
        const int i = nrow + threadIdx.y + j * 8;
        const int o = ocol + threadIdx.x;
        float v = 0.f;
        if (i < N_OUT) {
            float d = den[i];
            d = (d == 0.f) ? 1.f : d;
            v = num[(size_t)i * C + o] / d;
        }
        tile[threadIdx.y + j * 8][threadIdx.x] = v;
    }
    __syncthreads();
#pragma unroll
    for (int j = 0; j < 4; ++j) {
        const int o = ocol + threadIdx.y + j * 8;
        const int i = nrow + threadIdx.x;
        if (i < N_OUT) y[(size_t)o * N_OUT + i] = tile[threadIdx.x][threadIdx.y + j * 8];
    }
}

extern "C" void kernel_launch(void* const* d_in, const int* in_sizes, int n_in,
                              void* d_out, int out_size, void* d_ws, size_t ws_size,
                              hipStream_t stream) {
    const float* x     = (const float*)d_in[0];
    const int*   edges = (const int*)d_in[1];
    const float* W     = (const float*)d_in[2];
    const float* b     = (const float*)d_in[3];
    const float* att   = (const float*)d_in[4];
    const float* alpha = (const float*)d_in[5];
    float* out = (float*)d_out;

    float* ws  = (float*)d_ws;
    float* Wx  = ws + WX_OFF;
    float* s   = ws + S_OFF;
    float* num = ws + NUM_OFF;
    float* den = ws + DEN_OFF;

    // zero the accumulators (graph-capturable async memset)
    hipMemsetAsync(num, 0, (size_t)N_OUT * C * sizeof(float), stream);
    hipMemsetAsync(den, 0, (size_t)N_OUT * sizeof(float), stream);

    // Phase 1: WMMA GEMM, 64 nodes per block (4 waves x 16 nodes)
    gat_gemm_wx<<<(N_SRC + 63) / 64, 128, 0, stream>>>(x, W, b, Wx);

    // Phase 2a: per-node attention scalar, one wave per node
    gat_node_scalar<<<(N_SRC * 32 + 255) / 256, 256, 0, stream>>>(Wx, att, alpha, s);

    // Phase 2b: edge scatter, one wave per edge (8 edges per 256-thread block)
    gat_edge_scatter<<<(N_EDGES + 7) / 8, 256, 0, stream>>>(edges, Wx, s, num, den);

    // Phase 3: divide + transpose to (C_OUT, N_OUT)
    gat_divide_transpose<<<dim3(C / 32, (N_OUT + 31) / 32), dim3(32, 8), 0, stream>>>(num, den, out);
}